// GHM_Loss_11596411700016
// MI455X (gfx1250) — compile-verified
//
#include <hip/hip_runtime.h>
#include <hip/hip_bf16.h>

// GHM loss, single streaming pass:
//   mean(weight*bce) = sum_k S_k / max(cnt_k * nonempty, 1e-4)
// Kernel 1 accumulates 10 counts + 10 BCE sums (deterministic trees),
// Kernel 2 finalizes the scalar.

#ifndef USE_ASYNC_LDS
#define USE_ASYNC_LDS 1   // gfx1250 async global->LDS, 4-deep pipeline
#endif

#define NBLK  1024
#define NTHR  256
#define QUADS 4194304u                      // 16,777,216 elements / 4
#define ITERS (QUADS / (NBLK * NTHR))       // = 16 exactly, no tail

// ---------------- gfx1250 async-to-LDS helpers (inline asm) ----------------
__device__ __forceinline__ unsigned lds_off32(const void* p) {
  // generic pointer to LDS: low 32 bits are the LDS byte offset
  return (unsigned)(unsigned long long)p;
}
__device__ __forceinline__ void async_load_b128(unsigned lds, const void* gaddr) {
  asm volatile("global_load_async_to_lds_b128 %0, %1, off"
               :: "v"(lds), "v"(gaddr) : "memory");
}
__device__ __forceinline__ void async_load_b64(unsigned lds, const void* gaddr) {
  asm volatile("global_load_async_to_lds_b64 %0, %1, off"
               :: "v"(lds), "v"(gaddr) : "memory");
}
template <int N>
__device__ __forceinline__ void wait_async() {
  asm volatile("s_wait_asynccnt %0" :: "i"(N) : "memory");
}
__device__ __forceinline__ void wait_ds_0() {
  asm volatile("s_wait_dscnt 0x0" ::: "memory");
}

// ---------------- per-element GHM math ----------------
// t = exp(-|x|); e = sigma(-|x|) = t/(1+t) in (0, 0.5]
//   g = e        if (y==1) == (x>=0)     (bins 0..4)
//   g = 1 - e    otherwise               (bins 4..9)
// bce = max(x,0) - x*y + log(1+t)
__device__ __forceinline__ void ghm_elem(float xv, int y,
                                         float* __restrict__ sums,
                                         unsigned long long& cA,
                                         unsigned long long& cB) {
  float ax = fabsf(xv);
  float t  = __expf(-ax);
  float e  = t * __builtin_amdgcn_rcpf(1.0f + t);
  bool  match = ((y != 0) == (xv >= 0.0f));
  float g  = match ? e : (1.0f - e);
  int   b  = (int)(g * 9.9999f);          // floor; g in (0,1) -> b in [0,9]
  float bce = fmaxf(xv, 0.0f) - (y ? xv : 0.0f) + __logf(1.0f + t);

  // packed counters: bins 0..4 in cA, 5..9 in cB, 12-bit fields (max 64/thread)
  int f = (b >= 5) ? (b - 5) : b;
  unsigned long long inc = 1ull << (12 * f);
  cA += (b < 5) ? inc : 0ull;
  cB += (b < 5) ? 0ull : inc;

#pragma unroll
  for (int k = 0; k < 10; ++k)
    sums[k] += (b == k) ? bce : 0.0f;
}

__device__ __forceinline__ void consume_quad(float4 xv, int2 tv,
                                             float* __restrict__ sums,
                                             unsigned long long& cA,
                                             unsigned long long& cB) {
  ghm_elem(xv.x, (tv.x == 0), sums, cA, cB);
  ghm_elem(xv.y, (tv.x == 1), sums, cA, cB);
  ghm_elem(xv.z, (tv.y == 0), sums, cA, cB);
  ghm_elem(xv.w, (tv.y == 1), sums, cA, cB);
}

// ---------------- kernel 1: streaming pass ----------------
__global__ void __launch_bounds__(NTHR)
ghm_main(const float4* __restrict__ x4, const int2* __restrict__ t2,
         float* __restrict__ partials) {
  const int tid = threadIdx.x;
  unsigned q = blockIdx.x * NTHR + tid;
  const unsigned stride = NBLK * NTHR;

  float sums[10];
#pragma unroll
  for (int k = 0; k < 10; ++k) sums[k] = 0.0f;
  unsigned long long cA = 0ull, cB = 0ull;

#if USE_ASYNC_LDS
  // 4-deep double-buffer ring: ~3 KB async bytes in flight per wave
  __shared__ float4 xb[4][NTHR];
  __shared__ int2   tb[4][NTHR];

  // prologue: issue pairs 0,1,2
#pragma unroll
  for (int p = 0; p < 3; ++p) {
    async_load_b128(lds_off32(&xb[p][tid]), (const void*)(x4 + q + p * stride));
    async_load_b64 (lds_off32(&tb[p][tid]), (const void*)(t2 + q + p * stride));
  }

  for (int it = 0; it < ITERS - 3; ++it) {
    // buffer (it+3)&3 was last read at iteration it-1; ensure reads are done
    wait_ds_0();
    int nb = (it + 3) & 3;
    async_load_b128(lds_off32(&xb[nb][tid]), (const void*)(x4 + q + 3 * stride));
    async_load_b64 (lds_off32(&tb[nb][tid]), (const void*)(t2 + q + 3 * stride));
    wait_async<6>();                       // pair 'it' completed, 3 pairs pending
    int cb = it & 3;
    consume_quad(xb[cb][tid], tb[cb][tid], sums, cA, cB);
    q += stride;
  }
  // epilogue: drain pairs ITERS-3, ITERS-2, ITERS-1
  wait_async<4>();
  consume_quad(xb[(ITERS - 3) & 3][tid], tb[(ITERS - 3) & 3][tid], sums, cA, cB);
  wait_async<2>();
  consume_quad(xb[(ITERS - 2) & 3][tid], tb[(ITERS - 2) & 3][tid], sums, cA, cB);
  wait_async<0>();
  consume_quad(xb[(ITERS - 1) & 3][tid], tb[(ITERS - 1) & 3][tid], sums, cA, cB);
#else
#pragma unroll 4
  for (int it = 0; it < ITERS; ++it) {
    consume_quad(x4[q], t2[q], sums, cA, cB);
    q += stride;
  }
#endif

  // unpack counters (exact integers in f32: per-thread max 64, block max 16384)
  float vals[20];
#pragma unroll
  for (int k = 0; k < 5; ++k) vals[k]     = (float)((unsigned)(cA >> (12 * k)) & 0xFFFu);
#pragma unroll
  for (int k = 0; k < 5; ++k) vals[5 + k] = (float)((unsigned)(cB >> (12 * k)) & 0xFFFu);
#pragma unroll
  for (int k = 0; k < 10; ++k) vals[10 + k] = sums[k];

  // deterministic wave32 shfl-down tree for all 20 slots
#pragma unroll
  for (int k = 0; k < 20; ++k) {
    float v = vals[k];
#pragma unroll
    for (int off = 16; off > 0; off >>= 1) v += __shfl_down(v, off, 32);
    vals[k] = v;
  }

  __shared__ float red[NTHR / 32][20];
  int lane = tid & 31, wv = tid >> 5;
  if (lane == 0) {
#pragma unroll
    for (int k = 0; k < 20; ++k) red[wv][k] = vals[k];
  }
  __syncthreads();
  if (tid < 20) {
    float s = 0.0f;
#pragma unroll
    for (int w = 0; w < NTHR / 32; ++w) s += red[w][tid];   // fixed order
    partials[blockIdx.x * 20 + tid] = s;
  }
}

// ---------------- kernel 2: finalize ----------------
// partials: [NBLK][20]  slots 0..9 = counts, 10..19 = bce sums
__global__ void __launch_bounds__(NTHR)
ghm_finalize(const float* __restrict__ partials, float* __restrict__ out) {
  __shared__ float acc[20][8];
  __shared__ float fin[20];
  const int t = threadIdx.x;
  if (t < 160) {
    int k = t % 20, c = t / 20;              // 8 chunks of 128 block-partials
    float s = 0.0f;
    int base = c * (NBLK / 8);
    for (int p = 0; p < NBLK / 8; ++p) s += partials[(base + p) * 20 + k];
    acc[k][c] = s;
  }
  __syncthreads();
  if (t < 20) {
    float s = 0.0f;
#pragma unroll
    for (int c = 0; c < 8; ++c) s += acc[t][c];  // fixed order
    fin[t] = s;
  }
  __syncthreads();
  if (t == 0) {
    float nonempty = 0.0f;
#pragma unroll
    for (int k = 0; k < 10; ++k) nonempty += (fin[k] > 0.0f) ? 1.0f : 0.0f;
    float res = 0.0f;
#pragma unroll
    for (int k = 0; k < 10; ++k)
      res += fin[10 + k] / fmaxf(fin[k] * nonempty, 1e-4f);
    out[0] = res;   // == mean(weight * bce)
  }
}

extern "C" void kernel_launch(void* const* d_in, const int* in_sizes, int n_in,
                              void* d_out, int out_size, void* d_ws, size_t ws_size,
                              hipStream_t stream) {
  (void)in_sizes; (void)n_in; (void)out_size; (void)ws_size;
  const float4* x4 = (const float4*)d_in[0];   // x: [8388608, 2] f32, row-major
  const int2*   t2 = (const int2*)d_in[1];     // target: [8388608] int
  float* partials  = (float*)d_ws;             // NBLK*20 floats = 80 KB
  float* out       = (float*)d_out;

  ghm_main<<<NBLK, NTHR, 0, stream>>>(x4, t2, partials);
  ghm_finalize<<<1, NTHR, 0, stream>>>(partials, out);
}